// Network_31121333027489
// MI455X (gfx1250) — compile-verified
//
#include <hip/hip_runtime.h>
#include <stdint.h>

// ---------------------------------------------------------------------------
// FDTD single step, 4096x4096 fp32. Memory-bound (~402 MB min traffic ->
// ~17us at 23.3 TB/s). Fused single pass with halo recomputation:
//   stage Hx/Hy 72x72 tiles into LDS via TDM (tensor_load_to_lds),
//   compute Enew(68x68) with V_WMMA_F32_16X16X4_F32 banded-matmul convs,
//   emit E/Hx/Hy 64x64 outputs with float4 stores.
// Outer 8-wide ring handled by a small scalar frame kernel.
// ---------------------------------------------------------------------------

#define NXg 4096
#define NYg 4096
#define TS  64
#define HALO 4
#define LT  72              /* TS + 2*HALO */
#define FAST_LO 8
#define FAST_HI (NXg - 8)   /* 4088 */

typedef float        v2f __attribute__((ext_vector_type(2)));
typedef float        v8f __attribute__((ext_vector_type(8)));
typedef unsigned int v4u __attribute__((ext_vector_type(4)));
typedef int          v4i __attribute__((ext_vector_type(4)));
typedef int          v8i __attribute__((ext_vector_type(8)));

// filt = [a, -w, w, -a], a = (w-1)/3
static __device__ __forceinline__ float coef4(int t, float w, float a) {
    if (t == 0) return a;
    if (t == 1) return -w;
    if (t == 2) return w;
    if (t == 3) return -a;
    return 0.0f;
}

// ---------------------------------------------------------------------------
// Fast interior kernel: outputs in [8,4088) x [8,4088).
// ---------------------------------------------------------------------------
__global__ void __launch_bounds__(256)
fdtd_fast(const float* __restrict__ E0, const float* __restrict__ Hx0,
          const float* __restrict__ Hy0, const float* __restrict__ Wp,
          float* __restrict__ outE, float* __restrict__ outHx,
          float* __restrict__ outHy)
{
    __shared__ float smHx[LT * LT];
    __shared__ float smHy[LT * LT];
    __shared__ float smEn[LT * LT];

    const float w = Wp[0];
    const float a = (w - 1.0f) * (1.0f / 3.0f);
    const float C = 4095.0f / 400.0f;   // Z*DT/DX == DT/(Z*DY), Z=1, DX=DY

    const int ox = FAST_LO + blockIdx.x * TS;
    const int oy = FAST_LO + blockIdx.y * TS;
    const int wlenx = min(FAST_HI - ox, TS);
    const int wleny = min(FAST_HI - oy, TS);
    const int lx0 = min(ox - HALO, NXg - LT);   // load-window origin (in-bounds)
    const int ly0 = min(oy - HALO, NYg - LT);
    const int di = ox - lx0;                    // 4 normally, 16 on last tile
    const int dj = oy - ly0;

    const int tid  = threadIdx.x;
    const int lane = tid & 31;
    // Scalarized wave id: forces SGPR compares / scalar branches, so the TDM
    // issue below happens exactly once per designated wave and the WMMA tile
    // loop runs with EXEC = all ones by construction.
    const int wvu  = __builtin_amdgcn_readfirstlane(tid >> 5);

    // ---- Stage 0: stage Hx0 / Hy0 72x72 tiles into LDS -------------------
#if __has_builtin(__builtin_amdgcn_tensor_load_to_lds)
    if (wvu < 2) {   // waves 0 and 1 each issue one TDM 2-D tile load
        const float* src = (wvu == 0) ? Hx0 : Hy0;
        const uint32_t dstOff =
            (uint32_t)(uintptr_t)((wvu == 0) ? &smHx[0] : &smHy[0]);
        const uint64_t ga =
            (uint64_t)(uintptr_t)(src + (size_t)lx0 * NYg + ly0);

        v4u g0;
        g0.x = 1u;                                     // count=1 (valid D#)
        g0.y = dstOff;                                 // lds_addr (bytes)
        g0.z = (uint32_t)ga;                           // global_addr[31:0]
        g0.w = (uint32_t)((ga >> 32) & 0x01FFFFFFu)    // global_addr[56:32]
             | (2u << 30);                             // type=2 ("image")

        v8i g1;
        g1[0] = (int)(2u << 16);                                   // data_size=4B
        g1[1] = (int)(((uint32_t)NYg & 0xFFFFu) << 16);            // tensor_dim0.lo
        g1[2] = (int)((((uint32_t)NYg >> 16) & 0xFFFFu)            // tensor_dim0.hi
                      | (((uint32_t)NXg & 0xFFFFu) << 16));        // tensor_dim1.lo
        g1[3] = (int)((((uint32_t)NXg >> 16) & 0xFFFFu)            // tensor_dim1.hi
                      | ((uint32_t)LT << 16));                     // tile_dim0 = 72
        g1[4] = LT;                                                // tile_dim1 = 72
        g1[5] = NYg;                                               // dim0_stride lo
        g1[6] = 0;
        g1[7] = 0;

        v4i gz = {0, 0, 0, 0};
#if __clang_major__ >= 23
        v8i gz8 = {0, 0, 0, 0, 0, 0, 0, 0};
        __builtin_amdgcn_tensor_load_to_lds(g0, g1, gz, gz, gz8, 0);
#else
        __builtin_amdgcn_tensor_load_to_lds(g0, g1, gz, gz, 0);
#endif
#if __has_builtin(__builtin_amdgcn_s_wait_tensorcnt)
        __builtin_amdgcn_s_wait_tensorcnt(0);
#else
        asm volatile("s_wait_tensorcnt 0x0" ::: "memory");
#endif
    }
#else
    // Fallback: cooperative float4 copies.
    for (int k = tid; k < 2 * (LT * (LT / 4)); k += 256) {
        const int arr = k / (LT * (LT / 4));
        const int t2  = k % (LT * (LT / 4));
        const int r   = t2 / (LT / 4);
        const int c4  = (t2 % (LT / 4)) * 4;
        const float* src = arr ? Hy0 : Hx0;
        float*       dst = arr ? smHy : smHx;
        const float4 v =
            *(const float4*)(src + (size_t)(lx0 + r) * NYg + (ly0 + c4));
        *(float4*)(dst + r * LT + c4) = v;
    }
#endif
    __syncthreads();

    // ---- Stage 1: Enew on local [2,70)^2 ---------------------------------
    // Enew = E0 + C*( conv_x(filt,Hy) - conv_y(filt,Hx) )
#if __has_builtin(__builtin_amdgcn_wmma_f32_16x16x4_f32)
    {
        const int lh = lane >> 4;     // lane half (selects K pair / M half)
        const int mm = lane & 15;     // M (A,D) or N (B,D)

        // Banded coefficient operand, per documented f32 16x4 A layout:
        // lane<16: VGPRs hold K=4c+0,4c+1 ; lane>=16: K=4c+2,4c+3 ; M=lane&15.
        // G[m,j] = filt[j-m]; same values serve (negated) as F[j,n]=filt[j-n].
        v2f gA[5];
#pragma unroll
        for (int c = 0; c < 5; ++c) {
            gA[c].x = coef4(4 * c + 2 * lh + 0 - mm, w, a);
            gA[c].y = coef4(4 * c + 2 * lh + 1 - mm, w, a);
        }

        const int oset[5] = {2, 18, 34, 50, 54};   // covers [2,70), overlap ok
        for (int t = wvu; t < 25; t += 8) {        // scalar loop: EXEC stays full
            const int ti = oset[t / 5];
            const int tj = oset[t % 5];
            v8f acc = {0.f, 0.f, 0.f, 0.f, 0.f, 0.f, 0.f, 0.f};

            // +conv_x: acc += G(16x4 chunk) x Hy(4x16 chunk)
#pragma unroll
            for (int c = 0; c < 5; ++c) {
                const int rb = ti - 2 + 4 * c + 2 * lh;
                v2f B;
                B.x = smHy[rb * LT + tj + mm];
                B.y = smHy[(rb + 1) * LT + tj + mm];
                acc = __builtin_amdgcn_wmma_f32_16x16x4_f32(
                    false, gA[c], false, B, (short)0, acc, false, false);
            }
            // -conv_y: acc += Hx(16x4 chunk) x (-F)(4x16 chunk)
#pragma unroll
            for (int c = 0; c < 5; ++c) {
                const int cb = tj - 2 + 4 * c + 2 * lh;
                v2f A;
                A.x = smHx[(ti + mm) * LT + cb];
                A.y = smHx[(ti + mm) * LT + cb + 1];
                v2f Bn;
                Bn.x = -gA[c].x;
                Bn.y = -gA[c].y;
                acc = __builtin_amdgcn_wmma_f32_16x16x4_f32(
                    false, A, false, Bn, (short)0, acc, false, false);
            }
            // D layout: VGPR r -> M = r (lanes 0-15) / r+8 (lanes 16-31), N = lane&15
#pragma unroll
            for (int r = 0; r < 8; ++r) {
                const int row = ti + r + 8 * lh;
                const int col = tj + mm;
                const float e0 = E0[(lx0 + row) * NYg + (ly0 + col)];
                smEn[row * LT + col] = fmaf(C, acc[r], e0);
            }
        }
    }
#else
    for (int k = tid; k < 68 * 68; k += 256) {
        const int i = 2 + k / 68;
        const int j = 2 + k % 68;
        const float s1 = a * smHy[(i - 2) * LT + j] - w * smHy[(i - 1) * LT + j]
                       + w * smHy[i * LT + j]       - a * smHy[(i + 1) * LT + j];
        const float s2 = a * smHx[i * LT + j - 2] - w * smHx[i * LT + j - 1]
                       + w * smHx[i * LT + j]     - a * smHx[i * LT + j + 1];
        smEn[i * LT + j] = E0[(lx0 + i) * NYg + (ly0 + j)] + C * (s1 - s2);
    }
#endif
    __syncthreads();

    // ---- Stage 2: E / Hx / Hy outputs, float4 stores ---------------------
    const int rr = tid >> 4;           // 0..15
    const int cc = (tid & 15) << 2;    // 0,4,...,60
    if (cc < wleny) {
        for (int pass = 0; pass < 4; ++pass) {
            const int orow = pass * 16 + rr;
            if (orow < wlenx) {
                const int i  = di + orow;
                const int jb = dj + cc;
                float e4[4], hx4[4], hy4[4];
#pragma unroll
                for (int u = 0; u < 4; ++u) {
                    const int j = jb + u;
                    const float en = smEn[i * LT + j];
                    const float sy = a * smEn[i * LT + j - 1] - w * en
                                   + w * smEn[i * LT + j + 1]
                                   - a * smEn[i * LT + j + 2];
                    const float sx = a * smEn[(i - 1) * LT + j] - w * en
                                   + w * smEn[(i + 1) * LT + j]
                                   - a * smEn[(i + 2) * LT + j];
                    e4[u]  = en;
                    hx4[u] = smHx[i * LT + j] - C * sy;
                    hy4[u] = smHy[i * LT + j] + C * sx;
                }
                const size_t gp = (size_t)(ox + orow) * NYg + (oy + cc);
                *(float4*)(outE  + gp) = make_float4(e4[0],  e4[1],  e4[2],  e4[3]);
                *(float4*)(outHx + gp) = make_float4(hx4[0], hx4[1], hx4[2], hx4[3]);
                *(float4*)(outHy + gp) = make_float4(hy4[0], hy4[1], hy4[2], hy4[3]);
            }
        }
    }
}

// ---------------------------------------------------------------------------
// Frame kernel: the outer ring (p<8 || p>=4088 || q<8 || q>=4088).
// ---------------------------------------------------------------------------
static __device__ __forceinline__ float
enew_at(const float* __restrict__ E0, const float* __restrict__ Hx0,
        const float* __restrict__ Hy0, int p, int q, float w, float a, float C)
{
    const float e = E0[p * NYg + q];
    if (p <= 0 || p >= NXg - 1 || q <= 0 || q >= NYg - 1) return e;  // copy
    if (p >= 2 && p < NXg - 2 && q >= 2 && q < NYg - 2) {            // filt
        const float s1 = a * Hy0[(p - 2) * NYg + q] - w * Hy0[(p - 1) * NYg + q]
                       + w * Hy0[p * NYg + q]       - a * Hy0[(p + 1) * NYg + q];
        const float s2 = a * Hx0[p * NYg + q - 2] - w * Hx0[p * NYg + q - 1]
                       + w * Hx0[p * NYg + q]     - a * Hx0[p * NYg + q + 1];
        return e + C * (s1 - s2);
    }
    // boundary 2-tap [-1,1]
    const float s1 = Hy0[p * NYg + q] - Hy0[(p - 1) * NYg + q];
    const float s2 = Hx0[p * NYg + q] - Hx0[p * NYg + q - 1];
    return e + C * (s1 - s2);
}

__global__ void __launch_bounds__(256)
fdtd_frame(const float* __restrict__ E0, const float* __restrict__ Hx0,
           const float* __restrict__ Hy0, const float* __restrict__ Wp,
           float* __restrict__ outE, float* __restrict__ outHx,
           float* __restrict__ outHy)
{
    const int idx   = blockIdx.x * 256 + threadIdx.x;
    const int n_tb  = 8 * NYg;              // top/bottom strips
    const int n_all = 2 * n_tb + (NXg - 16) * 16;
    if (idx >= n_all) return;

    int p, q;
    if (idx < n_tb)            { p = idx / NYg;                 q = idx % NYg; }
    else if (idx < 2 * n_tb)   { const int t = idx - n_tb;
                                 p = (NXg - 8) + t / NYg;       q = t % NYg; }
    else                       { const int t = idx - 2 * n_tb;
                                 p = 8 + (t >> 4);
                                 const int c = t & 15;
                                 q = (c < 8) ? c : (NYg - 16 + c); }

    const float w = Wp[0];
    const float a = (w - 1.0f) * (1.0f / 3.0f);
    const float C = 4095.0f / 400.0f;

    const float e = enew_at(E0, Hx0, Hy0, p, q, w, a, C);

    float hx;
    if (p >= 2 && p < NXg - 2 && q >= 1 && q < NYg - 2) {        // filt
        const float s = a * enew_at(E0, Hx0, Hy0, p, q - 1, w, a, C) - w * e
                      + w * enew_at(E0, Hx0, Hy0, p, q + 1, w, a, C)
                      - a * enew_at(E0, Hx0, Hy0, p, q + 2, w, a, C);
        hx = Hx0[p * NYg + q] - C * s;
    } else if (p >= 1 && p < NXg - 1 && q < NYg - 1) {           // bc
        hx = Hx0[p * NYg + q]
           - C * (enew_at(E0, Hx0, Hy0, p, q + 1, w, a, C) - e);
    } else {
        hx = Hx0[p * NYg + q];
    }

    float hy;
    if (p >= 1 && p < NXg - 2 && q >= 2 && q < NYg - 2) {        // filt
        const float s = a * enew_at(E0, Hx0, Hy0, p - 1, q, w, a, C) - w * e
                      + w * enew_at(E0, Hx0, Hy0, p + 1, q, w, a, C)
                      - a * enew_at(E0, Hx0, Hy0, p + 2, q, w, a, C);
        hy = Hy0[p * NYg + q] + C * s;
    } else if (p < NXg - 1 && q >= 1 && q < NYg - 1) {           // bc
        hy = Hy0[p * NYg + q]
           + C * (enew_at(E0, Hx0, Hy0, p + 1, q, w, a, C) - e);
    } else {
        hy = Hy0[p * NYg + q];
    }

    outE [p * NYg + q] = e;
    outHx[p * NYg + q] = hx;
    outHy[p * NYg + q] = hy;
}

// ---------------------------------------------------------------------------
extern "C" void kernel_launch(void* const* d_in, const int* in_sizes, int n_in,
                              void* d_out, int out_size, void* d_ws,
                              size_t ws_size, hipStream_t stream)
{
    (void)in_sizes; (void)n_in; (void)d_ws; (void)ws_size; (void)out_size;

    const float* E0  = (const float*)d_in[0];
    const float* Hx0 = (const float*)d_in[1];
    const float* Hy0 = (const float*)d_in[2];
    const float* Wp  = (const float*)d_in[3];

    float* outE  = (float*)d_out;
    float* outHx = outE + (size_t)NXg * NYg;
    float* outHy = outHx + (size_t)NXg * NYg;

    dim3 grid((FAST_HI - FAST_LO + TS - 1) / TS,
              (FAST_HI - FAST_LO + TS - 1) / TS);   // 64 x 64
    fdtd_fast<<<grid, 256, 0, stream>>>(E0, Hx0, Hy0, Wp, outE, outHx, outHy);

    const int n_frame = 2 * (8 * NYg) + (NXg - 16) * 16;   // 130816
    fdtd_frame<<<(n_frame + 255) / 256, 256, 0, stream>>>(E0, Hx0, Hy0, Wp,
                                                          outE, outHx, outHy);
}